// EMDBlock_15436112462088
// MI455X (gfx1250) — compile-verified
//
#include <hip/hip_runtime.h>

// EMD residual: 32768 independent length-1024 f32 signals, 4 modes x 10 sifts.
// One wave32 per signal. The working signal lives entirely in registers
// (32 elements/lane); prev/next-extremum indices AND values are carried by
// wave-level (index,value) pair scans, so the sift inner loop does ZERO LDS
// traffic. LDS is used only to stage the initial load via the gfx1250 async
// global->LDS copy path (ASYNCcnt + s_wait_asynccnt).

#define SIG_LEN 1024
#define EPL 32            // elements per lane (1024 / 32 lanes)
#define WPB 8             // waves (signals) per block
#define N_MODES 4
#define N_SIFTS 10

// exclusive (idx,val) max-by-index scan across the wave; identity (-1, 0)
__device__ __forceinline__ void wscan_pair_excl_max(int& i, float& v, int lane) {
#pragma unroll
  for (int d = 1; d < 32; d <<= 1) {
    int ui = __shfl_up(i, d, 32);
    float uv = __shfl_up(v, d, 32);
    if (lane >= d && ui > i) { i = ui; v = uv; }
  }
  int ei = __shfl_up(i, 1, 32);
  float ev = __shfl_up(v, 1, 32);
  i = (lane == 0) ? -1 : ei;
  v = (lane == 0) ? 0.0f : ev;
}

// reverse exclusive (idx,val) min-by-index scan; identity (SIG_LEN, 0)
__device__ __forceinline__ void wscan_pair_excl_min_rev(int& i, float& v, int lane) {
#pragma unroll
  for (int d = 1; d < 32; d <<= 1) {
    int ui = __shfl_down(i, d, 32);
    float uv = __shfl_down(v, d, 32);
    if (lane < 32 - d && ui < i) { i = ui; v = uv; }
  }
  int ei = __shfl_down(i, 1, 32);
  float ev = __shfl_down(v, 1, 32);
  i = (lane == 31) ? SIG_LEN : ei;
  v = (lane == 31) ? 0.0f : ev;
}

__global__ __launch_bounds__(WPB * 32) void emd_residual_kernel(
    const float* __restrict__ q, float* __restrict__ out, int nsig) {
  __shared__ float sh[WPB][SIG_LEN];

  const int lane = threadIdx.x & 31;
  const int wv = threadIdx.x >> 5;
  const int sid = blockIdx.x * WPB + wv;
  if (sid >= nsig) return;

  float* S = &sh[wv][0];
  const float* gq = q + (size_t)sid * SIG_LEN;

  // ---- gfx1250 async copy: 4 KB signal global -> LDS (8 x 512B per wave) ----
  unsigned ldsAddr = (unsigned)(unsigned long long)(const void*)S + (unsigned)lane * 16u;
  unsigned goff = (unsigned)lane * 16u;
#pragma unroll
  for (int k = 0; k < 8; ++k) {
    asm volatile("global_load_async_to_lds_b128 %0, %1, %2"
                 :: "v"(ldsAddr + (unsigned)k * 512u),
                    "v"(goff + (unsigned)k * 512u),
                    "s"(gq)
                 : "memory");
  }
  asm volatile("s_wait_asynccnt 0x0" ::: "memory");
  asm volatile("" ::: "memory");

  const int base = lane * EPL;

  // load my contiguous 32-element chunk from LDS (vector b128 reads)
  float s0[EPL];  // working signal h (register-resident across all sifts)
  float r[EPL];   // running residual
  {
    const float4* S4 = (const float4*)(S + base);
#pragma unroll
    for (int k = 0; k < EPL / 4; ++k) {
      float4 t = S4[k];
      s0[4 * k + 0] = t.x; s0[4 * k + 1] = t.y;
      s0[4 * k + 2] = t.z; s0[4 * k + 3] = t.w;
    }
  }
#pragma unroll
  for (int j = 0; j < EPL; ++j) r[j] = s0[j];

#pragma unroll 1
  for (int mode = 0; mode < N_MODES; ++mode) {
#pragma unroll 1
    for (int it = 0; it < N_SIFTS; ++it) {
      // ---- chunk-boundary neighbors via shuffles (no LDS) ----
      float leftN = __shfl_up(s0[EPL - 1], 1, 32);   // s[base-1]
      float rightN = __shfl_down(s0[0], 1, 32);      // s[base+EPL]

      // ---- extrema masks (bitfields); endpoints forced extrema ----
      unsigned mmax = 0u, mmin = 0u;
#pragma unroll
      for (int j = 0; j < EPL; ++j) {
        float sm1 = (j == 0) ? leftN : s0[j - 1];
        float sp1 = (j == EPL - 1) ? rightN : s0[j + 1];
        int i = base + j;
        bool edge = (i == 0) | (i == SIG_LEN - 1);
        bool mx = edge | ((s0[j] > sm1) & (s0[j] > sp1));
        bool mn = edge | ((s0[j] < sm1) & (s0[j] < sp1));
        mmax |= ((unsigned)mx) << j;
        mmin |= ((unsigned)mn) << j;
      }

      // ---- pass A: lane-local last/first extremum (index AND value) ----
      int liM = -1, liN = -1, fiM = SIG_LEN, fiN = SIG_LEN;
      float lvM = 0.0f, lvN = 0.0f, fvM = 0.0f, fvN = 0.0f;
#pragma unroll
      for (int j = 0; j < EPL; ++j) {
        int i = base + j;
        if ((mmax >> j) & 1u) { liM = i; lvM = s0[j]; }
        if ((mmin >> j) & 1u) { liN = i; lvN = s0[j]; }
      }
#pragma unroll
      for (int j = EPL - 1; j >= 0; --j) {
        int i = base + j;
        if ((mmax >> j) & 1u) { fiM = i; fvM = s0[j]; }
        if ((mmin >> j) & 1u) { fiN = i; fvN = s0[j]; }
      }

      // ---- cross-lane carries: (idx,val) pair scans ----
      int cMi = liM, cNi = liN;  float cMv = lvM, cNv = lvN;
      wscan_pair_excl_max(cMi, cMv, lane);
      wscan_pair_excl_max(cNi, cNv, lane);
      int dMi = fiM, dNi = fiN;  float dMv = fvM, dNv = fvN;
      wscan_pair_excl_min_rev(dMi, dMv, lane);
      wscan_pair_excl_min_rev(dNi, dNv, lane);

      // ---- pass B forward: prev index (packed) + prev value per element ----
      unsigned pp[EPL];
      float vM[EPL], vN[EPL];
      {
        int pm = cMi, pn = cNi;
        float vm = cMv, vn = cNv;
#pragma unroll
        for (int j = 0; j < EPL; ++j) {
          int i = base + j;
          if ((mmax >> j) & 1u) { pm = i; vm = s0[j]; }
          if ((mmin >> j) & 1u) { pn = i; vn = s0[j]; }
          pp[j] = (unsigned)pm | ((unsigned)pn << 16);
          vM[j] = vm;
          vN[j] = vn;
        }
      }

      // ---- pass B backward: next (idx,val) on the fly; envelopes; update ----
      {
        int nm = dMi, nn = dNi;
        float wm = dMv, wn = dNv;
#pragma unroll
        for (int j = EPL - 1; j >= 0; --j) {
          int i = base + j;
          if ((mmax >> j) & 1u) { nm = i; wm = s0[j]; }
          if ((mmin >> j) & 1u) { nn = i; wn = s0[j]; }
          int pmj = (int)(pp[j] & 0xffffu);
          int pnj = (int)(pp[j] >> 16);
          // t = (i-prev)/(next-prev) when next>prev else 0 (v_rcp + mul)
          float tM = (nm > pmj)
                         ? (float)(i - pmj) * __builtin_amdgcn_rcpf((float)(nm - pmj))
                         : 0.0f;
          float tN = (nn > pnj)
                         ? (float)(i - pnj) * __builtin_amdgcn_rcpf((float)(nn - pnj))
                         : 0.0f;
          float envM = vM[j] + tM * (wm - vM[j]);
          float envN = vN[j] + tN * (wn - vN[j]);
          s0[j] = s0[j] - 0.5f * (envM + envN);
        }
      }
    }

    // r -= extracted IMF; working signal for next mode is the new r
#pragma unroll
    for (int j = 0; j < EPL; ++j) {
      r[j] = r[j] - s0[j];
      s0[j] = r[j];
    }
  }

  // ---- write residual (contiguous 128B per lane -> b128 stores) ----
  float* go = out + (size_t)sid * SIG_LEN + base;
#pragma unroll
  for (int j = 0; j < EPL; ++j) go[j] = r[j];
}

extern "C" void kernel_launch(void* const* d_in, const int* in_sizes, int n_in,
                              void* d_out, int out_size, void* d_ws, size_t ws_size,
                              hipStream_t stream) {
  const float* q = (const float*)d_in[0];  // (32,1024,8,128) f32; k/v/weight unused
  float* out = (float*)d_out;
  int nsig = in_sizes[0] / SIG_LEN;        // 32768 signals of length 1024
  int blocks = (nsig + WPB - 1) / WPB;     // 8 signals (waves) per block
  emd_residual_kernel<<<blocks, WPB * 32, 0, stream>>>(q, out, nsig);
}